// KVCache_3753801417331
// MI455X (gfx1250) — compile-verified
//
#include <hip/hip_runtime.h>

// KV-cache scatter update for MI455X (gfx1250, wave32).
// Shapes (from reference): B=4, H=32, MAX_S=4096, D=128, S=in_sizes[0] (=512).
// d_out = [k_out (B,H,MAX_S,D) | v_out (B,H,MAX_S,D)] float32.
//
// Strategy: build inv[MAX_S] (position -> new-row index, or -1) in d_ws, then a
// single fused streaming pass writes every output float4 exactly once, sourcing
// from k/v (scattered rows) or k_cache/v_cache (untouched rows). Pure HBM-bound:
// ~1 GiB total traffic -> ~46 us at 23.3 TB/s. All accesses are 16B/lane
// (global_load_b128 / global_store_b128) with non-temporal cache hints for the
// big streams. Branch on inv[s] is wave32-uniform (s is constant across a wave).

typedef float __attribute__((ext_vector_type(4))) v4f;

static constexpr int kB     = 4;
static constexpr int kH     = 32;
static constexpr int kMaxS  = 4096;
static constexpr int kD     = 128;
static constexpr int kDV    = kD / 4;              // float4 per row = 32
static constexpr unsigned kRowShift = 5;           // log2(kDV)
static constexpr unsigned kBHShift  = 17;          // log2(kMaxS * kDV) = log2(131072)
static constexpr unsigned kNVec     = (unsigned)kB * kH * kMaxS * kDV; // 16,777,216

__global__ void kv_init_inv(int* __restrict__ inv) {
    inv[blockIdx.x * 256 + threadIdx.x] = -1;
}

__global__ void kv_fill_inv(const int* __restrict__ pos, int* __restrict__ inv, int S) {
    int i = blockIdx.x * 256 + threadIdx.x;
    if (i < S) inv[pos[i]] = i;
}

__global__ void kv_fused_write(const v4f* __restrict__ k,
                               const v4f* __restrict__ v,
                               const v4f* __restrict__ kc,
                               const v4f* __restrict__ vc,
                               const int* __restrict__ inv,
                               v4f* __restrict__ ko,
                               v4f* __restrict__ vo,
                               int S) {
    unsigned idx  = blockIdx.x * 256u + threadIdx.x;      // < kNVec, exact grid
    unsigned dvec = idx & (kDV - 1);                      // varies within wave
    unsigned s    = (idx >> kRowShift) & (kMaxS - 1);     // uniform per wave32
    unsigned bh   = idx >> kBHShift;                      // uniform per wave32

    int j = inv[s];                                       // cached, wave-uniform
    v4f kval, vval;
    if (j >= 0) {
        size_t src = ((size_t)bh * (unsigned)S + (unsigned)j) * kDV + dvec;
        kval = __builtin_nontemporal_load(k + src);
        vval = __builtin_nontemporal_load(v + src);
    } else {
        kval = __builtin_nontemporal_load(kc + idx);
        vval = __builtin_nontemporal_load(vc + idx);
    }
    __builtin_nontemporal_store(kval, ko + idx);
    __builtin_nontemporal_store(vval, vo + idx);
}

// ---- Fallback path (no workspace): full copy, then scatter overwrite ----

__global__ void kv_copy2(const v4f* __restrict__ kc, const v4f* __restrict__ vc,
                         v4f* __restrict__ ko, v4f* __restrict__ vo) {
    unsigned idx = blockIdx.x * 256u + threadIdx.x;
    __builtin_nontemporal_store(__builtin_nontemporal_load(kc + idx), ko + idx);
    __builtin_nontemporal_store(__builtin_nontemporal_load(vc + idx), vo + idx);
}

__global__ void kv_scatter(const v4f* __restrict__ k, const v4f* __restrict__ v,
                           const int* __restrict__ pos,
                           v4f* __restrict__ ko, v4f* __restrict__ vo,
                           int S, unsigned total) {
    unsigned idx = blockIdx.x * 256u + threadIdx.x;
    if (idx >= total) return;
    unsigned dvec  = idx & (kDV - 1);
    unsigned t     = idx >> kRowShift;
    unsigned s_new = t % (unsigned)S;                     // uniform per wave32
    unsigned bh    = t / (unsigned)S;
    int p = pos[s_new];
    size_t dst = ((size_t)bh * kMaxS + (unsigned)p) * kDV + dvec;
    ko[dst] = k[idx];
    vo[dst] = v[idx];
}

extern "C" void kernel_launch(void* const* d_in, const int* in_sizes, int n_in,
                              void* d_out, int out_size, void* d_ws, size_t ws_size,
                              hipStream_t stream) {
    const int* input_pos = (const int*)d_in[0];
    const v4f* k  = (const v4f*)d_in[1];
    const v4f* v  = (const v4f*)d_in[2];
    const v4f* kc = (const v4f*)d_in[3];
    const v4f* vc = (const v4f*)d_in[4];
    const int  S  = in_sizes[0];

    const size_t nFloats = (size_t)kB * kH * kMaxS * kD;  // per output array
    v4f* ko = (v4f*)d_out;
    v4f* vo = (v4f*)((float*)d_out + nFloats);

    if (ws_size >= (size_t)kMaxS * sizeof(int)) {
        // Optimal single-pass path: 1 GiB total traffic.
        int* inv = (int*)d_ws;
        kv_init_inv<<<kMaxS / 256, 256, 0, stream>>>(inv);
        kv_fill_inv<<<(S + 255) / 256, 256, 0, stream>>>(input_pos, inv, S);
        kv_fused_write<<<kNVec / 256, 256, 0, stream>>>(k, v, kc, vc, inv, ko, vo, S);
    } else {
        // Fallback: copy whole caches, then overwrite scattered rows.
        kv_copy2<<<kNVec / 256, 256, 0, stream>>>(kc, vc, ko, vo);
        unsigned nsc = (unsigned)kB * kH * (unsigned)S * kDV;
        kv_scatter<<<(nsc + 255) / 256, 256, 0, stream>>>(k, v, input_pos, ko, vo, S, nsc);
    }
}